// Upsample2d_77575699300874
// MI455X (gfx1250) — compile-verified
//
#include <hip/hip_runtime.h>
#include <math.h>

#define IC_N 512
#define OC_N 512
#define H_N 128
#define W_N 128
#define HW_N (H_N * W_N)
#define PH 129            // per-phase buffer spatial dim (max over phases)
#define PHS (PH * PH)     // 16641 floats per channel per phase

typedef __attribute__((ext_vector_type(2))) float v2f;
typedef __attribute__((ext_vector_type(8))) float v8f;

// ---------------------------------------------------------------------------
// Kernel 1: demodulate + scale + flip + phase-pack weights.
// wp layout (floats): W00 [OC][2048] | W01 [OC][1024] | W10 [OC][1024] | W11 [OC][512]
// K index inside a phase = tap*512 + ic.
// Stored weight for tap (ky,kx) is wt[ky][kx] = w[2-ky][2-kx] * scale * demod(oc).
// ---------------------------------------------------------------------------
__global__ __launch_bounds__(256) void prep_weights(const float* __restrict__ wgt,
                                                    float* __restrict__ wp) {
  const int oc = blockIdx.x;
  const int tid = threadIdx.x;
  const float scale = rsqrtf(4608.0f); // 1/sqrt(IC*K*K)
  __shared__ float red[256];
  __shared__ float dsh;

  const float* wo = wgt + (size_t)oc * IC_N * 9;
  float s = 0.f;
  for (int ic = tid; ic < IC_N; ic += 256) {
    const float* wi = wo + ic * 9;
#pragma unroll
    for (int k = 0; k < 9; ++k) {
      float v = scale * wi[k];
      s += v * v;
    }
  }
  red[tid] = s;
  __syncthreads();
  for (int off = 128; off > 0; off >>= 1) {
    if (tid < off) red[tid] += red[tid + off];
    __syncthreads();
  }
  if (tid == 0) dsh = rsqrtf(red[0] + 1e-8f);
  __syncthreads();
  const float d = dsh * scale;

  float* w00 = wp;
  float* w01 = wp + (size_t)OC_N * 2048;
  float* w10 = w01 + (size_t)OC_N * 1024;
  float* w11 = w10 + (size_t)OC_N * 1024;

  for (int ic = tid; ic < IC_N; ic += 256) {
    const float* wi = wo + ic * 9;
    // P00 taps: t0:(ky0,kx0)->w[2][2], t1:(0,2)->w[2][0], t2:(2,0)->w[0][2], t3:(2,2)->w[0][0]
    w00[(size_t)oc * 2048 + 0 * 512 + ic] = d * wi[2 * 3 + 2];
    w00[(size_t)oc * 2048 + 1 * 512 + ic] = d * wi[2 * 3 + 0];
    w00[(size_t)oc * 2048 + 2 * 512 + ic] = d * wi[0 * 3 + 2];
    w00[(size_t)oc * 2048 + 3 * 512 + ic] = d * wi[0 * 3 + 0];
    // P01 taps: t0:(0,1)->w[2][1], t1:(2,1)->w[0][1]
    w01[(size_t)oc * 1024 + 0 * 512 + ic] = d * wi[2 * 3 + 1];
    w01[(size_t)oc * 1024 + 1 * 512 + ic] = d * wi[0 * 3 + 1];
    // P10 taps: t0:(1,0)->w[1][2], t1:(1,2)->w[1][0]
    w10[(size_t)oc * 1024 + 0 * 512 + ic] = d * wi[1 * 3 + 2];
    w10[(size_t)oc * 1024 + 1 * 512 + ic] = d * wi[1 * 3 + 0];
    // P11 tap: (1,1)->w[1][1]
    w11[(size_t)oc * 512 + ic] = d * wi[1 * 3 + 1];
  }
}

// ---------------------------------------------------------------------------
// Kernel 2: one transpose-conv phase as a GEMM via V_WMMA_F32_16X16X4_F32.
// Block = 128 threads (4 waves). Each wave: one 16 OC x 16 px tile; the 4
// waves share the 16-pixel column window (64 OC total) so the x (B-matrix)
// chunk is staged once per block in LDS.
// dymask/dxmask: bit t set => input offset -1 for tap t.
// up buffer: [512 oc][129][129] per phase (coalesced stores).
// ---------------------------------------------------------------------------
__global__ __launch_bounds__(128) void upconv_gemm(
    const float* __restrict__ x, const float* __restrict__ wp,
    float* __restrict__ up, int Kp, int ntaps, int dymask, int dxmask,
    int rows, int cols) {
  __shared__ float bs[512]; // [px 0..15][ic_local 0..31]

  const int tid = threadIdx.x;
  const int lane = tid & 31;
  const int wave = tid >> 5;
  const int px = lane & 15;      // M index for A, N index for B/C
  const int hi = lane >> 4;      // 0 or 1
  const int koff = hi * 2;

  const int r = blockIdx.y;
  const int colbase = blockIdx.x * 16;
  const int oc_base = blockIdx.z * 64 + wave * 16;

  const int l_icl0 = tid >> 4; // 0..7  (staging thread mapping)
  const int l_px = tid & 15;

  v8f acc = {};

  for (int t = 0; t < ntaps; ++t) {
    const int dy = -((dymask >> t) & 1);
    const int dx = -((dxmask >> t) & 1);
    const int row_in = r + dy;
    if (row_in < 0 || row_in >= H_N) continue; // uniform over block

    const int cin = colbase + l_px + dx;
    const bool cok = (cin >= 0) && (cin < W_N);
    const float* xrow = x + (size_t)row_in * W_N + cin;
    const float* arow = wp + (size_t)(oc_base + px) * Kp + t * IC_N;

    for (int icc = 0; icc < IC_N; icc += 32) {
      // stage 32 channels x 16 pixels of x into LDS (coalesced over px)
#pragma unroll
      for (int i = 0; i < 4; ++i) {
        const int icl = l_icl0 + 8 * i;
        float v = cok ? xrow[(size_t)(icc + icl) * HW_N] : 0.0f;
        bs[l_px * 32 + icl] = v;
      }
      if (icc + 32 < IC_N) // warm L2/L0 for the next channel chunk
        __builtin_prefetch(xrow + (size_t)(icc + 32 + l_icl0) * HW_N, 0, 1);
      __syncthreads();

#pragma unroll
      for (int s = 0; s < 8; ++s) {
        v2f a = *(const v2f*)(arow + icc + 4 * s + koff);       // A[M=px][K=koff..]
        v2f b = *(const v2f*)(&bs[px * 32 + 4 * s + koff]);     // B[K=koff..][N=px]
        acc = __builtin_amdgcn_wmma_f32_16x16x4_f32(
            false, a, false, b, (short)0, acc, false, false);
      }
      __syncthreads();
    }
  }

  const int c_out = colbase + px;
  if (c_out < cols) {
    float* o = up + (size_t)(oc_base + hi * 8) * PHS + (size_t)r * PH + c_out;
#pragma unroll
    for (int j = 0; j < 8; ++j) o[(size_t)j * PHS] = acc[j]; // oc = base+hi*8+j
  }
}

// ---------------------------------------------------------------------------
// Kernel 3: 4x4 depthwise blur (pad 1) over the phase-separated 257x257
// intermediate, producing the final 512x256x256 output. LDS-tiled 19x19 halo.
// bk[i][j] = B[i]*B[j]/16 with B = {1,3,3,1}  (== 4 * outer(k1d,k1d)).
// ---------------------------------------------------------------------------
__global__ __launch_bounds__(256) void blur_kernel(const float* __restrict__ up,
                                                   float* __restrict__ out) {
  __shared__ float tile[19 * 20];
  const int oc = blockIdx.z;
  const int y0 = blockIdx.y * 16;
  const int x0 = blockIdx.x * 16;
  const int tid = threadIdx.x;

  for (int idx = tid; idx < 19 * 19; idx += 256) {
    const int ly = idx / 19, lx = idx % 19;
    const int yy = y0 - 1 + ly;
    const int xx = x0 - 1 + lx;
    float v = 0.f;
    if (yy >= 0 && yy <= 256 && xx >= 0 && xx <= 256) {
      const int p = ((yy & 1) << 1) | (xx & 1);
      v = up[((size_t)p * OC_N + oc) * PHS + (size_t)(yy >> 1) * PH + (xx >> 1)];
    }
    tile[ly * 20 + lx] = v;
  }
  __syncthreads();

  const int ty = tid >> 4, tx = tid & 15;
  const float c0 = 0.25f, c1 = 0.75f; // B[i]/4 so product = B[i]B[j]/16
  const float cy[4] = {c0, c1, c1, c0};
  float acc = 0.f;
#pragma unroll
  for (int i = 0; i < 4; ++i) {
    const float* row = &tile[(ty + i) * 20 + tx];
    const float rsum = c0 * row[0] + c1 * row[1] + c1 * row[2] + c0 * row[3];
    acc += cy[i] * rsum;
  }
  out[((size_t)oc * 256 + (y0 + ty)) * 256 + (x0 + tx)] = acc;
}

// ---------------------------------------------------------------------------
extern "C" void kernel_launch(void* const* d_in, const int* in_sizes, int n_in,
                              void* d_out, int out_size, void* d_ws, size_t ws_size,
                              hipStream_t stream) {
  const float* x = (const float*)d_in[0];    // (1,512,128,128)
  const float* wgt = (const float*)d_in[1];  // (1,512,512,3,3)
  float* out = (float*)d_out;                // (1,512,256,256)

  float* ws = (float*)d_ws;
  float* wp = ws;                                   // 2,359,296 floats (9 MB)
  float* up = ws + (size_t)OC_N * 4608;             // 4 * 512 * 129 * 129 floats (136 MB)

  const float* w00 = wp;
  const float* w01 = wp + (size_t)OC_N * 2048;
  const float* w10 = w01 + (size_t)OC_N * 1024;
  const float* w11 = w10 + (size_t)OC_N * 1024;

  float* up00 = up + (size_t)0 * OC_N * PHS;
  float* up01 = up + (size_t)1 * OC_N * PHS;
  float* up10 = up + (size_t)2 * OC_N * PHS;
  float* up11 = up + (size_t)3 * OC_N * PHS;

  prep_weights<<<OC_N, 256, 0, stream>>>(wgt, wp);

  // phase (oy parity a, ox parity b): rows = a?128:129, cols = b?128:129
  upconv_gemm<<<dim3(9, 129, 8), 128, 0, stream>>>(x, w00, up00, 2048, 4, 0x3, 0x5, 129, 129);
  upconv_gemm<<<dim3(8, 129, 8), 128, 0, stream>>>(x, w01, up01, 1024, 2, 0x1, 0x0, 129, 128);
  upconv_gemm<<<dim3(9, 128, 8), 128, 0, stream>>>(x, w10, up10, 1024, 2, 0x0, 0x1, 128, 129);
  upconv_gemm<<<dim3(8, 128, 8), 128, 0, stream>>>(x, w11, up11,  512, 1, 0x0, 0x0, 128, 128);

  blur_kernel<<<dim3(16, 16, OC_N), 256, 0, stream>>>(up, out);
}